// SAM2Mask_73409581023267
// MI455X (gfx1250) — compile-verified
//
#include <hip/hip_runtime.h>
#include <hip/hip_bf16.h>
#include <stdint.h>

// ---- problem constants (match reference) ----
#define S_      1024
#define P_      60
#define K_      10
#define NM_     800
#define H_      256
#define W_      256
#define FD_     512
#define NMASK_  5
#define TOPK_   10
#define RADIUS_     0.2f
#define MASK_THRES_ 0.2f
#define SIM_THRES_  0.5f
#define BW_     25      // 800 mask bits / 32 = 25 words per superpoint bitset

typedef __attribute__((ext_vector_type(2))) float v2f;
typedef __attribute__((ext_vector_type(8))) float v8f;

// ------------------------------------------------------------------
// K1: deterministic FPS (start at index 0) per superpoint -> reps [S,K,3]
// one thread per superpoint; dist array in LDS
// ------------------------------------------------------------------
__global__ void k_fps(const float* __restrict__ pcds, float* __restrict__ reps) {
    __shared__ float sdist[64][P_];
    int s = blockIdx.x * blockDim.x + threadIdx.x;
    if (s >= S_) return;
    float* dist = sdist[threadIdx.x];
    const float* base = pcds + (size_t)s * P_ * 3;
    float cx = base[0], cy = base[1], cz = base[2];
    for (int p = 0; p < P_; ++p) {
        float dx = base[p*3+0]-cx, dy = base[p*3+1]-cy, dz = base[p*3+2]-cz;
        dist[p] = dx*dx + dy*dy + dz*dz;
    }
    float* r = reps + (size_t)s * K_ * 3;
    r[0] = cx; r[1] = cy; r[2] = cz;
    for (int k = 1; k < K_; ++k) {
        int best = 0; float bv = dist[0];
        for (int p = 1; p < P_; ++p) {           // argmax, first-max tie-break (== jnp.argmax)
            float v = dist[p];
            if (v > bv) { bv = v; best = p; }
        }
        float nx = base[best*3+0], ny = base[best*3+1], nz = base[best*3+2];
        r[k*3+0] = nx; r[k*3+1] = ny; r[k*3+2] = nz;
        for (int p = 0; p < P_; ++p) {
            float dx = base[p*3+0]-nx, dy = base[p*3+1]-ny, dz = base[p*3+2]-nz;
            dist[p] = fminf(dist[p], dx*dx + dy*dy + dz*dz);
        }
    }
}

// ------------------------------------------------------------------
// K2: adj0[i][j] = (min_{k,l} dist(rep_i_k, rep_j_l) < R && i!=j) ? exp(-mind) : 0
// thread per (i,j); replicates sq_i + sq_j - 2dot clamp + sqrt(+1e-12)
// ------------------------------------------------------------------
__global__ void k_adj0(const float* __restrict__ reps, float* __restrict__ adj0) {
    int j = blockIdx.x * blockDim.x + threadIdx.x;
    int i = blockIdx.y;
    if (j >= S_) return;
    float ix[K_], iy[K_], iz[K_], iq[K_];
    const float* ri = reps + (size_t)i * K_ * 3;
    #pragma unroll
    for (int k = 0; k < K_; ++k) {
        ix[k] = ri[k*3+0]; iy[k] = ri[k*3+1]; iz[k] = ri[k*3+2];
        iq[k] = ix[k]*ix[k] + iy[k]*iy[k] + iz[k]*iz[k];
    }
    const float* rj = reps + (size_t)j * K_ * 3;
    float mind = 3.4e38f;
    for (int k = 0; k < K_; ++k) {
        float jx = rj[k*3+0], jy = rj[k*3+1], jz = rj[k*3+2];
        float jq = jx*jx + jy*jy + jz*jz;
        #pragma unroll
        for (int l = 0; l < K_; ++l) {
            float dot = ix[l]*jx + iy[l]*jy + iz[l]*jz;
            float d2  = fmaxf(iq[l] + jq - 2.0f*dot, 0.0f);
            mind = fminf(mind, sqrtf(d2 + 1e-12f));
        }
    }
    float a = ((mind < RADIUS_) && (i != j)) ? expf(-mind) : 0.0f;
    adj0[(size_t)i * S_ + j] = a;
}

// ------------------------------------------------------------------
// K3: frac[s][m] = (#visible points of s inside mask m) / (P + 1e-6)
// block = one s x 256 masks; (y,x,vis) staged in LDS; masks are L2-resident
// jax bool array assumed 1 byte/element
// ------------------------------------------------------------------
__global__ void k_frac(const uint8_t* __restrict__ masks, const int* __restrict__ mapping,
                       float* __restrict__ frac) {
    int s = blockIdx.y;
    int m = blockIdx.x * blockDim.x + threadIdx.x;
    __shared__ int sidx[P_];
    if (threadIdx.x < P_) {
        int n = s * P_ + threadIdx.x;
        int y = mapping[n*4+1], x = mapping[n*4+2], v = mapping[n*4+3];
        y = min(max(y, 0), H_-1); x = min(max(x, 0), W_-1);
        sidx[threadIdx.x] = (v == 1) ? (y * W_ + x) : -1;
    }
    __syncthreads();
    if (m >= NM_) return;
    const uint8_t* mb = masks + (size_t)m * (H_ * W_);
    int cnt = 0;
    for (int p = 0; p < P_; ++p) {
        int idx = sidx[p];
        if (idx >= 0) cnt += (mb[idx] != 0);
    }
    frac[(size_t)s * NM_ + m] = (float)cnt / ((float)P_ + 1e-6f);
}

// ------------------------------------------------------------------
// K4: per-mask keep top-10 superpoints (exact lax.top_k semantics:
// value desc, index asc tie-break); zero the rest in place
// ------------------------------------------------------------------
__global__ void k_topk(float* __restrict__ frac) {
    int m = blockIdx.x;
    __shared__ float v[S_];
    for (int s = threadIdx.x; s < S_; s += blockDim.x)
        v[s] = frac[(size_t)s * NM_ + m];
    __syncthreads();
    for (int s = threadIdx.x; s < S_; s += blockDim.x) {
        float x = v[s];
        int rank = 0;
        for (int t = 0; t < S_; ++t) {
            float y = v[t];
            rank += ((y > x) || (y == x && t < s)) ? 1 : 0;
        }
        if (rank >= TOPK_) frac[(size_t)s * NM_ + m] = 0.0f;
    }
}

// ------------------------------------------------------------------
// K5: contrib bitsets (frac >= 0.2) + conf[s] = clip(sum(f*score)/(sum f + 1e-6), 0)
// one wave32 per superpoint; lane w owns mask word w
// ------------------------------------------------------------------
__global__ void k_bits_conf(const float* __restrict__ frac, const float* __restrict__ scores,
                            unsigned* __restrict__ bits, float* __restrict__ conf) {
    int s = blockIdx.x;
    int lane = threadIdx.x;          // 0..31 (wave32)
    float wsum = 0.0f, fsum = 0.0f;
    unsigned word = 0;
    int mbase = lane * 32;
    for (int b = 0; b < 32; ++b) {
        int m = mbase + b;
        if (m < NM_) {
            float f = frac[(size_t)s * NM_ + m];
            wsum += f * scores[m];
            fsum += f;
            if (f >= MASK_THRES_) word |= (1u << b);
        }
    }
    if (lane < BW_) bits[s * BW_ + lane] = word;
    for (int off = 16; off > 0; off >>= 1) {
        wsum += __shfl_down(wsum, off);
        fsum += __shfl_down(fsum, off);
    }
    if (lane == 0) conf[s] = fmaxf(wsum / (fsum + 1e-6f), 0.0f);
}

// ------------------------------------------------------------------
// K6: feats[s] = L2-normalize(mean of 60 point features), [S,512]
// block per superpoint, thread owns 2 dims
// ------------------------------------------------------------------
__global__ void k_feats(const float* __restrict__ pf, float* __restrict__ feats) {
    int s = blockIdx.x;
    int t = threadIdx.x;             // 0..255
    __shared__ float red[256];
    float a0 = 0.0f, a1 = 0.0f;
    const float* base = pf + (size_t)s * P_ * FD_;
    for (int p = 0; p < P_; ++p) {
        a0 += base[(size_t)p * FD_ + t];
        a1 += base[(size_t)p * FD_ + t + 256];
    }
    float inv = 1.0f / ((float)P_ + 1e-6f);
    a0 *= inv; a1 *= inv;
    red[t] = a0*a0 + a1*a1;
    __syncthreads();
    for (int off = 128; off > 0; off >>= 1) {
        if (t < off) red[t] += red[t + off];
        __syncthreads();
    }
    float nrm = sqrtf(red[0]);
    float sc  = 1.0f / fmaxf(nrm, 1e-12f);
    feats[(size_t)s * FD_ + t]       = a0 * sc;
    feats[(size_t)s * FD_ + t + 256] = a1 * sc;
}

// ------------------------------------------------------------------
// K7: G = feats @ feats^T  (1024x1024x512 f32) via V_WMMA_F32_16X16X4_F32
// one wave per 16x16 tile; 8 waves/block cover 128(M)x16(N); grid (8,64)
// A layout (ISA 7.12.2, 32-bit 16x4): lane<16 -> (M=lane, K=k..k+1),
// lane>=16 -> (M=lane-16, K=k+2..k+3). B mirrors with N in lanes.
// ------------------------------------------------------------------
__global__ void __launch_bounds__(256) k_gemm(const float* __restrict__ feats,
                                              float* __restrict__ G) {
    int lane = threadIdx.x & 31;
    int wave = threadIdx.x >> 5;
    int mbase = (blockIdx.x * 8 + wave) * 16;
    int nbase = blockIdx.y * 16;
    int half = lane >> 4;            // 0: K=0..1 | 1: K=2..3
    int l    = lane & 15;
    const float* arow = feats + (size_t)(mbase + l) * FD_ + half * 2;
    const float* brow = feats + (size_t)(nbase + l) * FD_ + half * 2;
    v8f c = {};
    #pragma unroll 4
    for (int k = 0; k < FD_; k += 4) {
        v2f a = *(const v2f*)(arow + k);   // 8B aligned: k%4==0, half*2 even
        v2f b = *(const v2f*)(brow + k);
        c = __builtin_amdgcn_wmma_f32_16x16x4_f32(
                /*neg_a=*/false, a, /*neg_b=*/false, b,
                /*c_mod=*/(short)0, c, /*reuse_a=*/false, /*reuse_b=*/false);
    }
    // C/D layout: VGPR r -> (M = r + 8*half, N = l)
    float* out = G + (size_t)(mbase + half * 8) * S_ + nbase + l;
    #pragma unroll
    for (int r = 0; r < 8; ++r) out[(size_t)r * S_] = c[r];
}

// ------------------------------------------------------------------
// K8: adjacency[i][j] = adj0>0 ? max(ov*sim', adj0) : 0
// ov from first-5-set-bits of (mi|cj) bitsets; sim' from GEMM result
// ------------------------------------------------------------------
__global__ void k_final(const float* __restrict__ adj0, const float* __restrict__ G,
                        const unsigned* __restrict__ bits, float* __restrict__ out) {
    int j = blockIdx.x * blockDim.x + threadIdx.x;
    int i = blockIdx.y;
    if (j >= S_) return;
    size_t idx = (size_t)i * S_ + j;
    float a0 = adj0[idx];
    float r = 0.0f;
    if (a0 > 0.0f) {
        const unsigned* bi = bits + i * BW_;
        const unsigned* bj = bits + j * BW_;
        int taken = 0, inter = 0, total = 0;
        for (int w = 0; w < BW_; ++w) {
            unsigned U = bi[w] | bj[w];
            unsigned I = bi[w] & bj[w];
            total += __popc(U);
            while (U && taken < NMASK_) {        // first 5 set bits, ascending m
                unsigned b = U & (~U + 1u);
                if (I & b) ++inter;
                ++taken;
                U &= U - 1u;
            }
        }
        int uni = total < NMASK_ ? total : NMASK_;
        float ov = (float)inter / ((float)uni + 1e-6f);
        float g = G[idx];
        float sim = fminf(fmaxf((g + 1.0f) * 0.5f, 0.0f), 1.0f);
        if (sim < SIM_THRES_) sim = 1.0f;
        r = fmaxf(ov * sim, a0);
    }
    out[idx] = r;
}

// ------------------------------------------------------------------
extern "C" void kernel_launch(void* const* d_in, const int* in_sizes, int n_in,
                              void* d_out, int out_size, void* d_ws, size_t ws_size,
                              hipStream_t stream) {
    const float*   pcds    = (const float*)d_in[0];
    // d_in[1] spt_labels: unused (labels are contiguous blocks of P)
    const uint8_t* masks   = (const uint8_t*)d_in[2];  // jax bool -> 1 byte/elem
    const int*     mapping = (const int*)d_in[3];
    const float*   scores  = (const float*)d_in[4];
    const float*   pf      = (const float*)d_in[5];

    float* adjacency = (float*)d_out;                   // [S,S]
    float* conf      = (float*)d_out + (size_t)S_ * S_; // [S]

    float* ws = (float*)d_ws;
    float* reps  = ws;  ws += (size_t)S_ * K_ * 3;      //   30720 f
    float* adj0  = ws;  ws += (size_t)S_ * S_;          // 1048576 f
    float* frac  = ws;  ws += (size_t)S_ * NM_;         //  819200 f
    float* feats = ws;  ws += (size_t)S_ * FD_;         //  524288 f
    float* G     = ws;  ws += (size_t)S_ * S_;          // 1048576 f
    unsigned* bits = (unsigned*)ws;                     // S_*BW_ words (~100 KB)

    k_fps      <<<dim3(S_ / 64),  dim3(64),  0, stream>>>(pcds, reps);
    k_adj0     <<<dim3(4, S_),    dim3(256), 0, stream>>>(reps, adj0);
    k_frac     <<<dim3(4, S_),    dim3(256), 0, stream>>>(masks, mapping, frac);
    k_topk     <<<dim3(NM_),      dim3(256), 0, stream>>>(frac);
    k_bits_conf<<<dim3(S_),       dim3(32),  0, stream>>>(frac, scores, bits, conf);
    k_feats    <<<dim3(S_),       dim3(256), 0, stream>>>(pf, feats);
    k_gemm     <<<dim3(8, 64),    dim3(256), 0, stream>>>(feats, G);
    k_final    <<<dim3(4, S_),    dim3(256), 0, stream>>>(adj0, G, bits, adjacency);
}